// DownsampleBlock_AR_63067299775044
// MI455X (gfx1250) — compile-verified
//
#include <hip/hip_runtime.h>
#include <hip/hip_bf16.h>

typedef __attribute__((ext_vector_type(2))) float v2f;
typedef __attribute__((ext_vector_type(8))) float v8f;

// D = A(16x4 f32) * B(4x16 f32) + C, full fp32 precision (matches reference dtype)
#define WMMA_F32X4(a, b, c) \
  __builtin_amdgcn_wmma_f32_16x16x4_f32(false, (a), false, (b), (short)0, (c), false, false)

// ---------------------------------------------------------------------------
// Zero-pad x in H: XP[b, i, hp(0..65), w] ; hp==0 and hp==65 are zero rows.
// ---------------------------------------------------------------------------
__global__ __launch_bounds__(256)
void pad_x_kernel(const float* __restrict__ x, float* __restrict__ xp) {
  const int q = blockIdx.x * 256 + threadIdx.x;   // 8*64*66*32 = 1,081,344
  if (q >= 1081344) return;
  const int w = q & 31;
  const int r = q >> 5;
  const int hp = r % 66;
  const int bi = r / 66;                          // b*64 + i
  float v = 0.f;
  if (hp >= 1 && hp <= 64)
    v = x[((size_t)bi * 64 + (hp - 1)) * 32 + w];
  xp[q] = v;
}

// Zero the two H-pad rows of a [(b, hp, t), c] activation buffer (Hp rows).
__global__ __launch_bounds__(256)
void zero_pad_rows(float* __restrict__ buf, int Hp) {
  const int q = blockIdx.x * 256 + threadIdx.x;   // 8*2*32*128 = 65,536
  const int rr   = q & 4095;                      // (t, c)
  const int side = (q >> 12) & 1;
  const int b    = q >> 13;
  const int hp   = side ? (Hp - 1) : 0;
  buf[((size_t)(b * Hp + hp) << 12) + rr] = 0.f;
}

// ---------------------------------------------------------------------------
// GEMM1: ha[(b,h),(c,t)] = sum_k XP[(b,h),k] * W[k,(c,t)] + bias(c,t)
// M=512, N=4096, K=6144.  W column (c,t) is contiguous in Wt memory.
// Wave tile 16x64 (4 accumulators). Block = 8 waves = 32M x 256N.
// Branchless inner loop: 5 global_load_b64 + 4 WMMA per K=4 step.
// ---------------------------------------------------------------------------
__global__ __launch_bounds__(256)
void gemm1_kernel(const float* __restrict__ xp, const float* __restrict__ Wt,
                  const float* __restrict__ bt, float* __restrict__ ha) {
  const int lane   = threadIdx.x & 31;
  const int wave   = threadIdx.x >> 5;
  const int lane16 = lane & 15;
  const int half   = lane >> 4;          // 0: K0/K1 slots, 1: K2/K3 slots

  const int m0 = blockIdx.y * 32 + (wave >> 2) * 16;
  const int n0 = blockIdx.x * 256 + (wave & 3) * 64;

  const int m  = m0 + lane16;            // A-fragment row for this lane
  const int b  = m >> 6;                 // batch (uniform within tile)
  const int h  = m & 63;                 // H position

  // XP row (b, i, hp=h+d): offset b*135168 + i*2112 + (h+d)*32
  const float* aBase = xp + (size_t)b * 135168 + h * 32 + 2 * half;

  const float* colp[4];
  int ncol[4];
  #pragma unroll
  for (int j = 0; j < 4; ++j) {
    const int n = n0 + j * 16 + lane16;  // B-fragment column for this lane
    ncol[j] = n;
    const int c = n >> 5, t = n & 31;
    colp[j] = Wt + (size_t)t * 786432 + (size_t)c * 6144 + 2 * half;
  }

  v8f acc[4] = {v8f{}, v8f{}, v8f{}, v8f{}};

  int kOff = 0;                          // k = i*96 + d*32 + w0, walked in order
  for (int i = 0; i < 64; ++i) {
    const float* arow0 = aBase + i * 2112;
    #pragma unroll
    for (int d = 0; d < 3; ++d) {
      const float* arow = arow0 + d * 32;
      #pragma unroll
      for (int w0 = 0; w0 < 32; w0 += 4) {
        const v2f a  = *(const v2f*)(arow + w0);
        const v2f b0 = *(const v2f*)(colp[0] + kOff);
        const v2f b1 = *(const v2f*)(colp[1] + kOff);
        const v2f b2 = *(const v2f*)(colp[2] + kOff);
        const v2f b3 = *(const v2f*)(colp[3] + kOff);
        acc[0] = WMMA_F32X4(a, b0, acc[0]);
        acc[1] = WMMA_F32X4(a, b1, acc[1]);
        acc[2] = WMMA_F32X4(a, b2, acc[2]);
        acc[3] = WMMA_F32X4(a, b3, acc[3]);
        kOff += 4;
      }
    }
  }

  #pragma unroll
  for (int j = 0; j < 4; ++j) {
    const int n = ncol[j];
    const int c = n >> 5, t = n & 31;
    const float bias = bt[t * 128 + c];
    #pragma unroll
    for (int v = 0; v < 8; ++v) {
      const int mr = m0 + v + 8 * half;  // D layout: VGPR v -> row v (+8 hi half)
      ha[(size_t)mr * 4096 + n] = acc[j][v] + bias;
    }
  }
}

// ---------------------------------------------------------------------------
// 3x1 conv as GEMM: M=(b,ho,t)=8192, N=co=128, K=(dh,c')=384.
// Ain layout [(b, hp, t), c'] with H pre-padded (Hp rows, zero pad rows), so
// input row = ho*stride + dh and the loop is branchless.
// Bt layout [co][dh*128+c'].  Wave tile 16x64; block 64M x 128N.
// ---------------------------------------------------------------------------
__global__ __launch_bounds__(256)
void gemm_conv3x1(const float* __restrict__ Ain, const float* __restrict__ Bt,
                  float* __restrict__ Out, int Hp, int stride) {
  const int lane   = threadIdx.x & 31;
  const int wave   = threadIdx.x >> 5;
  const int lane16 = lane & 15;
  const int half   = lane >> 4;

  const int m0 = blockIdx.x * 64 + (wave >> 1) * 16;
  const int n0 = (wave & 1) * 64;

  const int p  = m0 + lane16;
  const int t  = p & 31;
  const int rb = p >> 5;
  const int ho = rb & 31;                // Hout = 32
  const int b  = rb >> 5;

  // row (b, ho*stride + dh): base + dh*4096
  const float* aBase = Ain + ((size_t)(b * Hp + ho * stride) * 32 + t) * 128 + 2 * half;

  const float* bcol[4];
  #pragma unroll
  for (int j = 0; j < 4; ++j)
    bcol[j] = Bt + (size_t)(n0 + j * 16 + lane16) * 384 + 2 * half;

  v8f acc[4] = {v8f{}, v8f{}, v8f{}, v8f{}};

  #pragma unroll
  for (int dh = 0; dh < 3; ++dh) {
    const float* arow = aBase + dh * 4096;
    const int koff = dh * 128;
    #pragma unroll 4
    for (int c0 = 0; c0 < 128; c0 += 4) {
      const v2f a  = *(const v2f*)(arow + c0);
      const v2f b0 = *(const v2f*)(bcol[0] + koff + c0);
      const v2f b1 = *(const v2f*)(bcol[1] + koff + c0);
      const v2f b2 = *(const v2f*)(bcol[2] + koff + c0);
      const v2f b3 = *(const v2f*)(bcol[3] + koff + c0);
      acc[0] = WMMA_F32X4(a, b0, acc[0]);
      acc[1] = WMMA_F32X4(a, b1, acc[1]);
      acc[2] = WMMA_F32X4(a, b2, acc[2]);
      acc[3] = WMMA_F32X4(a, b3, acc[3]);
    }
  }

  #pragma unroll
  for (int j = 0; j < 4; ++j) {
    const int co = n0 + j * 16 + lane16;
    #pragma unroll
    for (int v = 0; v < 8; ++v)
      Out[(size_t)(m0 + v + 8 * half) * 128 + co] = acc[j][v];
  }
}

// ---------------------------------------------------------------------------
// Shortcut 1x1 stride-(2,1) conv as GEMM: M=8192, N=128, K=64.
// A[p,i] = x[b, i, 2*ho, t]
// ---------------------------------------------------------------------------
__global__ __launch_bounds__(256)
void gemm_shortcut(const float* __restrict__ x, const float* __restrict__ w2a,
                   float* __restrict__ Out) {
  const int lane   = threadIdx.x & 31;
  const int wave   = threadIdx.x >> 5;
  const int lane16 = lane & 15;
  const int half   = lane >> 4;

  const int m0 = blockIdx.x * 64 + (wave >> 1) * 16;
  const int n0 = (wave & 1) * 64;

  const int p  = m0 + lane16;
  const int t  = p & 31;
  const int rb = p >> 5;
  const int ho = rb & 31;
  const int b  = rb >> 5;

  // x flat: ((b*64+i)*64 + 2*ho)*32 + t = b*131072 + i*2048 + ho*64 + t
  const float* abase = x + (size_t)b * 131072 + ho * 64 + t;

  const float* bcol[4];
  #pragma unroll
  for (int j = 0; j < 4; ++j)
    bcol[j] = w2a + (size_t)(n0 + j * 16 + lane16) * 64 + 2 * half;

  v8f acc[4] = {v8f{}, v8f{}, v8f{}, v8f{}};

  #pragma unroll
  for (int k0 = 0; k0 < 64; k0 += 4) {
    const int i = k0 + 2 * half;
    v2f a = { abase[(size_t)i * 2048], abase[(size_t)(i + 1) * 2048] };
    const v2f b0 = *(const v2f*)(bcol[0] + k0);
    const v2f b1 = *(const v2f*)(bcol[1] + k0);
    const v2f b2 = *(const v2f*)(bcol[2] + k0);
    const v2f b3 = *(const v2f*)(bcol[3] + k0);
    acc[0] = WMMA_F32X4(a, b0, acc[0]);
    acc[1] = WMMA_F32X4(a, b1, acc[1]);
    acc[2] = WMMA_F32X4(a, b2, acc[2]);
    acc[3] = WMMA_F32X4(a, b3, acc[3]);
  }

  #pragma unroll
  for (int j = 0; j < 4; ++j) {
    const int co = n0 + j * 16 + lane16;
    #pragma unroll
    for (int v = 0; v < 8; ++v)
      Out[(size_t)(m0 + v + 8 * half) * 128 + co] = acc[j][v];
  }
}

// ---------------------------------------------------------------------------
// BN stats for ha layout [(b,h), (c,t)]: channel c = cols c*32..c*32+31.
// One block per channel; biased variance; fold gamma/beta into scale/shift.
// ---------------------------------------------------------------------------
__global__ __launch_bounds__(256)
void stats_ha(const float* __restrict__ ha, const float* __restrict__ g,
              const float* __restrict__ be, float* __restrict__ scale,
              float* __restrict__ shift) {
  __shared__ float ssum[256], ssq[256];
  const int c = blockIdx.x, tid = threadIdx.x;
  float s = 0.f, q = 0.f;
  for (int j = tid; j < 16384; j += 256) {
    const int m = j >> 5, t = j & 31;
    const float v = ha[(size_t)m * 4096 + (c << 5) + t];
    s += v; q += v * v;
  }
  ssum[tid] = s; ssq[tid] = q; __syncthreads();
  for (int o = 128; o > 0; o >>= 1) {
    if (tid < o) { ssum[tid] += ssum[tid + o]; ssq[tid] += ssq[tid + o]; }
    __syncthreads();
  }
  if (tid == 0) {
    const float mean = ssum[0] * (1.f / 16384.f);
    const float var  = ssq[0] * (1.f / 16384.f) - mean * mean;
    const float sc   = g[c] * rsqrtf(var + 1e-5f);
    scale[c] = sc;
    shift[c] = be[c] - mean * sc;
  }
}

// BN stats for [p, co] layout tensors (8192 rows x 128 cols)
__global__ __launch_bounds__(256)
void stats_pc(const float* __restrict__ buf, const float* __restrict__ g,
              const float* __restrict__ be, float* __restrict__ scale,
              float* __restrict__ shift) {
  __shared__ float ssum[256], ssq[256];
  const int c = blockIdx.x, tid = threadIdx.x;
  float s = 0.f, q = 0.f;
  for (int j = tid; j < 8192; j += 256) {
    const float v = buf[(size_t)j * 128 + c];
    s += v; q += v * v;
  }
  ssum[tid] = s; ssq[tid] = q; __syncthreads();
  for (int o = 128; o > 0; o >>= 1) {
    if (tid < o) { ssum[tid] += ssum[tid + o]; ssq[tid] += ssq[tid + o]; }
    __syncthreads();
  }
  if (tid == 0) {
    const float mean = ssum[0] * (1.f / 8192.f);
    const float var  = ssq[0] * (1.f / 8192.f) - mean * mean;
    const float sc   = g[c] * rsqrtf(var + 1e-5f);
    scale[c] = sc;
    shift[c] = be[c] - mean * sc;
  }
}

// BN+ReLU + transpose ha[(b,h),(c,t)] -> padded [(b, h+1, t), c] (Hp=66)
__global__ __launch_bounds__(256)
void bnrelu_tr(const float* __restrict__ ha, const float* __restrict__ sc,
               const float* __restrict__ sh, float* __restrict__ out) {
  const int q = blockIdx.x * 256 + threadIdx.x;   // 2,097,152 threads
  const int c = q & 127, t = (q >> 7) & 31, m = q >> 12;
  const int b = m >> 6, h = m & 63;
  const float v = ha[(size_t)m * 4096 + (c << 5) + t];
  out[((size_t)((b * 66 + h + 1) * 32 + t) << 7) + c] = fmaxf(sc[c] * v + sh[c], 0.f);
}

// BN+ReLU [p,co] -> padded [(b, ho+1, t), co] (Hp=34)
__global__ __launch_bounds__(256)
void bnrelu_pad(const float* __restrict__ in, const float* __restrict__ sc,
                const float* __restrict__ sh, float* __restrict__ out) {
  const int q = blockIdx.x * 256 + threadIdx.x;   // 1,048,576 threads
  const int c = q & 127, t = (q >> 7) & 31, ho = (q >> 12) & 31, b = q >> 17;
  out[((size_t)((b * 34 + ho + 1) * 32 + t) << 7) + c] = fmaxf(sc[c] * in[q] + sh[c], 0.f);
}

// w[co, c', dh] (dh fastest, stride 3) -> out[co, dh, c'] (c' fastest)
__global__ __launch_bounds__(256)
void transpose_w(const float* __restrict__ w, float* __restrict__ out) {
  const int idx = blockIdx.x * 256 + threadIdx.x;
  if (idx >= 49152) return;
  const int co = idx / 384, r = idx % 384, dh = r >> 7, c = r & 127;
  out[idx] = w[co * 384 + c * 3 + dh];
}

// out[b,co,ho,t] = relu( bn1b(h2) + bn2a(hb) ), inputs in [p,co] layout
__global__ __launch_bounds__(256)
void final_add(const float* __restrict__ h2, const float* __restrict__ hb,
               const float* __restrict__ s1b, const float* __restrict__ o1b,
               const float* __restrict__ s2a, const float* __restrict__ o2a,
               float* __restrict__ out) {
  const int q = blockIdx.x * 256 + threadIdx.x;   // 1,048,576 threads
  const int t = q & 31, ho = (q >> 5) & 31, co = (q >> 10) & 127, b = q >> 17;
  const int p = (b * 32 + ho) * 32 + t;
  const float av = s1b[co] * h2[(size_t)p * 128 + co] + o1b[co];
  const float bv = s2a[co] * hb[(size_t)p * 128 + co] + o2a[co];
  out[q] = fmaxf(av + bv, 0.f);
}

// ---------------------------------------------------------------------------
extern "C" void kernel_launch(void* const* d_in, const int* in_sizes, int n_in,
                              void* d_out, int out_size, void* d_ws, size_t ws_size,
                              hipStream_t stream) {
  const float* x   = (const float*)d_in[0];
  const float* Wt  = (const float*)d_in[1];
  const float* bt  = (const float*)d_in[2];
  const float* w1a = (const float*)d_in[3];
  const float* w1b = (const float*)d_in[4];
  const float* w2a = (const float*)d_in[5];
  const float* g_t  = (const float*)d_in[6];
  const float* b_t  = (const float*)d_in[7];
  const float* g_1a = (const float*)d_in[8];
  const float* b_1a = (const float*)d_in[9];
  const float* g_1b = (const float*)d_in[10];
  const float* b_1b = (const float*)d_in[11];
  const float* g_2a = (const float*)d_in[12];
  const float* b_2a = (const float*)d_in[13];
  float* out = (float*)d_out;

  float* w = (float*)d_ws;
  float* HA    = w;                    // 512*4096            = 2,097,152
  float* HANP  = w + 2097152;          // 8*66*32*128         = 2,162,688 (H-padded)
  float* H1    = w + 4259840;          // 8192*128            = 1,048,576
  float* H1NP  = w + 5308416;          // 8*34*32*128         = 1,114,112 (H-padded)
  float* H2    = w + 6422528;          // 8192*128            = 1,048,576
  float* HB    = w + 7471104;          // 8192*128            = 1,048,576
  float* XP    = w + 8519680;          // 8*64*66*32          = 1,081,344 (H-padded x)
  float* BT1A  = w + 9601024;          // 128*384
  float* BT1B  = w + 9650176;          // 128*384
  float* ST    = w + 9699328;          // stats: 8 x 128
  float* sc_t = ST,       *sh_t = ST + 128;
  float* s1a  = ST + 256, *o1a  = ST + 384;
  float* s1b  = ST + 512, *o1b  = ST + 640;
  float* s2a  = ST + 768, *o2a  = ST + 896;

  // prep: zero-padded x, transposed conv weights, zeroed activation pad rows
  pad_x_kernel<<<4224, 256, 0, stream>>>(x, XP);
  transpose_w<<<192, 256, 0, stream>>>(w1a, BT1A);
  transpose_w<<<192, 256, 0, stream>>>(w1b, BT1B);
  zero_pad_rows<<<256, 256, 0, stream>>>(HANP, 66);
  zero_pad_rows<<<256, 256, 0, stream>>>(H1NP, 34);

  // big ragged-temporal-conv einsum as one GEMM (M=512,N=4096,K=6144)
  gemm1_kernel<<<dim3(16, 16), 256, 0, stream>>>(XP, Wt, bt, HA);

  // BN_t stats, then BN+ReLU fused with transpose into padded [(b,hp,t),c]
  stats_ha<<<128, 256, 0, stream>>>(HA, g_t, b_t, sc_t, sh_t);
  bnrelu_tr<<<8192, 256, 0, stream>>>(HA, sc_t, sh_t, HANP);

  // conv1a (3x1, stride 2) as GEMM
  gemm_conv3x1<<<128, 256, 0, stream>>>(HANP, BT1A, H1, 66, 2);
  stats_pc<<<128, 256, 0, stream>>>(H1, g_1a, b_1a, s1a, o1a);
  bnrelu_pad<<<4096, 256, 0, stream>>>(H1, s1a, o1a, H1NP);

  // conv1b (3x1, stride 1) as GEMM
  gemm_conv3x1<<<128, 256, 0, stream>>>(H1NP, BT1B, H2, 34, 1);
  stats_pc<<<128, 256, 0, stream>>>(H2, g_1b, b_1b, s1b, o1b);

  // shortcut 1x1 stride-(2,1) conv as GEMM
  gemm_shortcut<<<128, 256, 0, stream>>>(x, w2a, HB);
  stats_pc<<<128, 256, 0, stream>>>(HB, g_2a, b_2a, s2a, o2a);

  // final BN(h2) + BN(hb), add, relu -> NCHW output
  final_add<<<4096, 256, 0, stream>>>(H2, HB, s1b, o1b, s2a, o2a, out);
}